// GEON_MODEL_25494925869180
// MI455X (gfx1250) — compile-verified
//
#include <hip/hip_runtime.h>
#include <hip/hip_bf16.h>

#define NPTS 1024
#define BATCH 8
#define KNN 20
#define BN_EPS 1e-5f
#define NEG_INF (-3.402823e38f)

typedef __attribute__((ext_vector_type(2))) float v2f;
typedef __attribute__((ext_vector_type(8))) float v8f;

// -------------------------------------------------------------------------
// norms[b*N+n] = sum_c X[b,c,n]^2
// -------------------------------------------------------------------------
__global__ void dgcnn_norms_kernel(const float* __restrict__ X, float* __restrict__ norms,
                                   int C, long xBatch) {
    int t = blockIdx.x * blockDim.x + threadIdx.x;
    if (t >= BATCH * NPTS) return;
    int b = t / NPTS, n = t % NPTS;
    const float* Xb = X + (long)b * xBatch;
    float s = 0.f;
    for (int c = 0; c < C; ++c) { float x = Xb[(long)c * NPTS + n]; s += x * x; }
    norms[t] = s;
}

// -------------------------------------------------------------------------
// Fused Gram-strip GEMM (WMMA f32 16x16x4) + per-row top-20 selection.
// Each block: one batch b, one 16-row strip i0..i0+15 of S = X^T X.
// Templated on C so the 16xC A-strip fragments stay register-resident and
// are reused across all 64 j-tiles (2 loads per WMMA in the inner loop).
// 4 waves fill strip[16][1024] in LDS, then threads 0..15 each select
// top-20 of (2*S_ij - ||xj||^2) for one row.
// -------------------------------------------------------------------------
template <int C>
__global__ __launch_bounds__(128)
void dgcnn_knn_kernel(const float* __restrict__ X, const float* __restrict__ norms,
                      int* __restrict__ nbr, long xBatch) {
    __shared__ float strip[16 * NPTS];  // 64 KB
    const int b    = blockIdx.y;
    const int i0   = blockIdx.x * 16;
    const int wave = threadIdx.x >> 5;
    const int lane = threadIdx.x & 31;
    const int half = lane >> 4;         // 0: K{0,1} / 1: K{2,3} (ISA 16x4 A layout)
    const int m    = lane & 15;
    const float* Xb = X + (long)b * xBatch;

    // Preload all A fragments for this 16-row strip (C/2 VGPRs per lane).
    v2f afrag[C / 4];
#pragma unroll
    for (int ks = 0; ks < C / 4; ++ks) {
        const float* pk0 = Xb + (long)(ks * 4 + 2 * half) * NPTS;
        afrag[ks].x = pk0[i0 + m];
        afrag[ks].y = pk0[NPTS + i0 + m];
    }

    for (int jt = wave; jt < NPTS / 16; jt += 4) {
        const int j0 = jt * 16;
        v8f acc = {};
#pragma unroll
        for (int ks = 0; ks < C / 4; ++ks) {
            const float* pk0 = Xb + (long)(ks * 4 + 2 * half) * NPTS;
            v2f bv;
            bv.x = pk0[j0 + m];
            bv.y = pk0[NPTS + j0 + m];
            acc = __builtin_amdgcn_wmma_f32_16x16x4_f32(false, afrag[ks], false, bv,
                                                        (short)0, acc, false, false);
        }
#pragma unroll
        for (int v = 0; v < 8; ++v)
            strip[(v + 8 * half) * NPTS + j0 + m] = acc[v];
    }
    __syncthreads();

    if (threadIdx.x < 16) {
        const int row = threadIdx.x;
        const float* nrm = norms + b * NPTS;
        float best[KNN]; int bidx[KNN];
#pragma unroll
        for (int t = 0; t < KNN; ++t) { best[t] = NEG_INF; bidx[t] = 0; }
        for (int j = 0; j < NPTS; ++j) {
            // ordering-equivalent to -||xi-xj||^2 (row-constant -||xi||^2 dropped)
            float d = 2.0f * strip[row * NPTS + j] - nrm[j];
            if (d > best[KNN - 1]) {
                best[KNN - 1] = d; bidx[KNN - 1] = j;
#pragma unroll
                for (int t = KNN - 1; t > 0; --t) {
                    if (best[t] > best[t - 1]) {
                        float tv = best[t]; best[t] = best[t - 1]; best[t - 1] = tv;
                        int ti = bidx[t];  bidx[t] = bidx[t - 1];  bidx[t - 1] = ti;
                    }
                }
            }
        }
        int* out = nbr + ((long)b * NPTS + (i0 + row)) * KNN;
#pragma unroll
        for (int t = 0; t < KNN; ++t) out[t] = bidx[t];
    }
}

// -------------------------------------------------------------------------
// Wq[o,c] = W[o,c]; Wp[o,c] = W[o,C+c] - W[o,c]   (W is O x 2C row-major)
// -------------------------------------------------------------------------
__global__ void dgcnn_prepw_kernel(const float* __restrict__ W,
                                   float* __restrict__ Wq, float* __restrict__ Wp,
                                   int O, int C) {
    int t = blockIdx.x * blockDim.x + threadIdx.x;
    if (t >= O * C) return;
    int o = t / C, c = t % C;
    float wq = W[(long)o * 2 * C + c];
    Wq[t] = wq;
    Wp[t] = W[(long)o * 2 * C + C + c] - wq;
}

// -------------------------------------------------------------------------
// Y[b,o,n] = sum_c A[o,c] * X[b,c,n]  via WMMA f32 16x16x4.
// One wave per 16x64 output tile: A fragment (float2 b64 load) amortized
// over 4 WMMAs per k-step.
// -------------------------------------------------------------------------
__global__ __launch_bounds__(32)
void dgcnn_gemm16_kernel(const float* __restrict__ A,   // O x C row-major
                         const float* __restrict__ X,   // per-batch C x NPTS
                         float* __restrict__ Y,         // per-batch O x NPTS
                         int C, long xBatch, long yBatch) {
    const int b    = blockIdx.z;
    const int o0   = blockIdx.y * 16;
    const int n0   = blockIdx.x * 64;
    const int lane = threadIdx.x & 31;
    const int half = lane >> 4;
    const int m    = lane & 15;
    const float* Xb = X + (long)b * xBatch;

    v8f acc0 = {}, acc1 = {}, acc2 = {}, acc3 = {};
    for (int k0 = 0; k0 < C; k0 += 4) {
        const float2 ap = *(const float2*)(A + (long)(o0 + m) * C + k0 + 2 * half);
        v2f av; av.x = ap.x; av.y = ap.y;
        const float* pk = Xb + (long)(k0 + 2 * half) * NPTS;
        v2f b0, b1, b2, b3;
        b0.x = pk[n0 + m];       b0.y = pk[NPTS + n0 + m];
        b1.x = pk[n0 + 16 + m];  b1.y = pk[NPTS + n0 + 16 + m];
        b2.x = pk[n0 + 32 + m];  b2.y = pk[NPTS + n0 + 32 + m];
        b3.x = pk[n0 + 48 + m];  b3.y = pk[NPTS + n0 + 48 + m];
        __builtin_prefetch(pk + 4 * NPTS + n0 + m, 0, 1);
        acc0 = __builtin_amdgcn_wmma_f32_16x16x4_f32(false, av, false, b0, (short)0, acc0, false, false);
        acc1 = __builtin_amdgcn_wmma_f32_16x16x4_f32(false, av, false, b1, (short)0, acc1, false, false);
        acc2 = __builtin_amdgcn_wmma_f32_16x16x4_f32(false, av, false, b2, (short)0, acc2, false, false);
        acc3 = __builtin_amdgcn_wmma_f32_16x16x4_f32(false, av, false, b3, (short)0, acc3, false, false);
    }
    float* Yb = Y + (long)b * yBatch;
#pragma unroll
    for (int v = 0; v < 8; ++v) {
        long r = (long)(o0 + v + 8 * half) * NPTS;
        Yb[r + n0 + m]      = acc0[v];
        Yb[r + n0 + 16 + m] = acc1[v];
        Yb[r + n0 + 32 + m] = acc2[v];
        Yb[r + n0 + 48 + m] = acc3[v];
    }
}

// -------------------------------------------------------------------------
// out[b,o,n] = relu(inv_o*(P[b,o,n] + max_k Q[b,o,idx[b,n,k]]) + beta_o)
// (BN gamma > 0 and ReLU are monotone => max commutes with BN+ReLU)
// -------------------------------------------------------------------------
__global__ void dgcnn_edgemax_kernel(const float* __restrict__ Q, const float* __restrict__ P,
                                     const int* __restrict__ nbr,
                                     const float* __restrict__ g, const float* __restrict__ be,
                                     const float* __restrict__ mu, const float* __restrict__ va,
                                     float* __restrict__ out, int O, long outBatch, int total) {
    int t = blockIdx.x * blockDim.x + threadIdx.x;
    if (t >= total) return;
    int n = t % NPTS;
    int o = (t / NPTS) % O;
    int b = t / (NPTS * O);
    float inv  = g[o] * rsqrtf(va[o] + BN_EPS);
    float beta = be[o] - mu[o] * inv;
    const int*   id   = nbr + ((long)b * NPTS + n) * KNN;
    const float* Qrow = Q + ((long)b * O + o) * NPTS;
    float mx = NEG_INF;
#pragma unroll
    for (int k = 0; k < KNN; ++k) mx = fmaxf(mx, Qrow[id[k]]);
    float val = inv * (P[((long)b * O + o) * NPTS + n] + mx) + beta;
    out[(long)b * outBatch + (long)o * NPTS + n] = fmaxf(val, 0.f);
}

// -------------------------------------------------------------------------
// pooled[b,o] = relu(inv_o * max_n H[b,o,n] + beta_o)
// -------------------------------------------------------------------------
__global__ void dgcnn_pool_kernel(const float* __restrict__ H,
                                  const float* __restrict__ g, const float* __restrict__ be,
                                  const float* __restrict__ mu, const float* __restrict__ va,
                                  float* __restrict__ pooled) {
    int t = blockIdx.x * blockDim.x + threadIdx.x;
    if (t >= BATCH * 1024) return;
    int o = t % 1024, b = t / 1024;
    const float* row = H + ((long)b * 1024 + o) * NPTS;
    float mx = NEG_INF;
    for (int n = 0; n < NPTS; ++n) mx = fmaxf(mx, row[n]);
    float inv = g[o] * rsqrtf(va[o] + BN_EPS);
    pooled[t] = fmaxf(inv * mx + be[o] - mu[o] * inv, 0.f);
}

// -------------------------------------------------------------------------
// out[b,o] = (optional relu)((optional bn)(in[b,:] . W[o,:] + bias[o]))
// -------------------------------------------------------------------------
__global__ void dgcnn_fc_kernel(const float* __restrict__ in, const float* __restrict__ W,
                                const float* __restrict__ bias,
                                const float* __restrict__ g, const float* __restrict__ be,
                                const float* __restrict__ mu, const float* __restrict__ va,
                                float* __restrict__ out, int I, int Onum, int doRelu) {
    int t = blockIdx.x * blockDim.x + threadIdx.x;
    if (t >= BATCH * Onum) return;
    int o = t % Onum, b = t / Onum;
    const float* row = in + (long)b * I;
    const float* w   = W + (long)o * I;
    float s = 0.f;
    for (int i = 0; i < I; ++i) s += row[i] * w[i];
    if (bias) s += bias[o];
    if (g) {
        float inv = g[o] * rsqrtf(va[o] + BN_EPS);
        s = inv * s + be[o] - mu[o] * inv;
    }
    out[t] = doRelu ? fmaxf(s, 0.f) : s;
}

// -------------------------------------------------------------------------
extern "C" void kernel_launch(void* const* d_in, const int* in_sizes, int n_in,
                              void* d_out, int out_size, void* d_ws, size_t ws_size,
                              hipStream_t stream) {
    const int B = BATCH, N = NPTS;
    const float* x   = (const float*)d_in[0];
    const float* W1  = (const float*)d_in[1];
    const float* W2  = (const float*)d_in[2];
    const float* W3  = (const float*)d_in[3];
    const float* W4  = (const float*)d_in[4];
    const float* W5  = (const float*)d_in[5];
    // bn params: groups of 4 starting at index 6
    const float* bn[7][4];
    for (int i = 0; i < 7; ++i)
        for (int j = 0; j < 4; ++j) bn[i][j] = (const float*)d_in[6 + 4 * i + j];
    const float* L1W = (const float*)d_in[34];
    const float* L2W = (const float*)d_in[35];
    const float* L2b = (const float*)d_in[36];
    const float* L3W = (const float*)d_in[37];
    const float* L3b = (const float*)d_in[38];

    // ---- workspace carve-up (floats) ----
    float* cat    = (float*)d_ws;                        // 8*512*1024
    float* Q      = cat + (size_t)B * 512 * N;           // 8*256*1024
    float* P      = Q   + (size_t)B * 256 * N;           // 8*256*1024
    float* H      = P   + (size_t)B * 256 * N;           // 8*1024*1024
    float* norms  = H   + (size_t)B * 1024 * N;          // 8*1024
    int*   nbr    = (int*)(norms + (size_t)B * N);       // 8*1024*20
    float* Wq     = (float*)(nbr + (size_t)B * N * KNN); // 256*128
    float* Wp     = Wq + 256 * 128;                      // 256*128
    float* pooled = Wp + 256 * 128;                      // 8*1024
    float* h1     = pooled + (size_t)B * 1024;           // 8*512
    float* h2     = h1 + (size_t)B * 512;                // 8*256

    auto run_block = [&](const float* Xi, long xBatch, int C, int O, const float* W,
                         const float* const* bnp, float* outBase) {
        dgcnn_norms_kernel<<<(B * N + 255) / 256, 256, 0, stream>>>(Xi, norms, C, xBatch);
        dim3 kg(N / 16, B);
        if (C == 4)
            dgcnn_knn_kernel<4><<<kg, 128, 0, stream>>>(Xi, norms, nbr, xBatch);
        else if (C == 64)
            dgcnn_knn_kernel<64><<<kg, 128, 0, stream>>>(Xi, norms, nbr, xBatch);
        else
            dgcnn_knn_kernel<128><<<kg, 128, 0, stream>>>(Xi, norms, nbr, xBatch);
        dgcnn_prepw_kernel<<<(O * C + 255) / 256, 256, 0, stream>>>(W, Wq, Wp, O, C);
        dgcnn_gemm16_kernel<<<dim3(N / 64, O / 16, B), 32, 0, stream>>>(Wq, Xi, Q, C, xBatch, (long)O * N);
        dgcnn_gemm16_kernel<<<dim3(N / 64, O / 16, B), 32, 0, stream>>>(Wp, Xi, P, C, xBatch, (long)O * N);
        int total = B * O * N;
        dgcnn_edgemax_kernel<<<(total + 255) / 256, 256, 0, stream>>>(
            Q, P, nbr, bnp[0], bnp[1], bnp[2], bnp[3], outBase, O, (long)512 * N, total);
    };

    // edge blocks write directly into their channel slice of cat (B,512,N)
    run_block(x,                   4L * N,   4,   64,  W1, bn[0], cat);
    run_block(cat,                 512L * N, 64,  64,  W2, bn[1], cat + (long)64 * N);
    run_block(cat + (long)64 * N,  512L * N, 64,  128, W3, bn[2], cat + (long)128 * N);
    run_block(cat + (long)128 * N, 512L * N, 128, 256, W4, bn[3], cat + (long)256 * N);

    // conv5: H[b,1024,N] = W5 (1024x512) @ cat
    dgcnn_gemm16_kernel<<<dim3(N / 64, 1024 / 16, B), 32, 0, stream>>>(
        W5, cat, H, 512, 512L * N, 1024L * N);
    dgcnn_pool_kernel<<<(B * 1024 + 255) / 256, 256, 0, stream>>>(
        H, bn[4][0], bn[4][1], bn[4][2], bn[4][3], pooled);

    // MLP head
    dgcnn_fc_kernel<<<(B * 512 + 255) / 256, 256, 0, stream>>>(
        pooled, L1W, nullptr, bn[5][0], bn[5][1], bn[5][2], bn[5][3], h1, 1024, 512, 1);
    dgcnn_fc_kernel<<<(B * 256 + 255) / 256, 256, 0, stream>>>(
        h1, L2W, L2b, bn[6][0], bn[6][1], bn[6][2], bn[6][3], h2, 512, 256, 1);
    dgcnn_fc_kernel<<<1, 64, 0, stream>>>(
        h2, L3W, L3b, nullptr, nullptr, nullptr, nullptr, (float*)d_out, 256, 8, 0);
}